// CentroidModule_36112085025109
// MI455X (gfx1250) — compile-verified
//
#include <hip/hip_runtime.h>

typedef __attribute__((ext_vector_type(2))) float v2f;
typedef __attribute__((ext_vector_type(4))) float v4f;
typedef __attribute__((ext_vector_type(8))) float v8f;
typedef __attribute__((ext_vector_type(4))) int   v4i;

// Address-space-typed int4 pointers for the async global->LDS builtin:
// param0 = global (AS1) source, param1 = LDS (AS3) destination.
typedef __attribute__((address_space(1))) v4i g_v4i;
typedef __attribute__((address_space(3))) v4i l_v4i;

constexpr int Bv = 64, Tv = 1024, Dv = 256, Kv = 512;
constexpr int Mv      = Bv * Tv;      // 65536 rows
constexpr int M_BLK   = 128;          // rows per workgroup (8 waves x 16 rows)
constexpr int NTILES  = Kv / 16;      // 32 centroid tiles
constexpr int KSTEPS  = Dv / 4;       // 64 k-steps of the 16x16x4 f32 WMMA
constexpr int PSTRIDE = 260;          // LDS row stride (floats): conflict-free, 16B-aligned

// ---------------------------------------------------------------------------
// ||c_k||^2 per centroid: one wave per row, shuffle reduction.
// ---------------------------------------------------------------------------
__global__ void csq_kernel(const float* __restrict__ protos, float* __restrict__ csq) {
    int lane = threadIdx.x & 31;
    int k = blockIdx.x * (blockDim.x >> 5) + (threadIdx.x >> 5);
    if (k >= Kv) return;
    const float* row = protos + (size_t)k * Dv;
    float s = 0.0f;
    #pragma unroll
    for (int j = 0; j < Dv / 32; ++j) {
        float v = row[lane + j * 32];
        s += v * v;
    }
    #pragma unroll
    for (int m = 16; m; m >>= 1) s += __shfl_xor(s, m, 32);
    if (lane == 0) csq[k] = s;
}

// ---------------------------------------------------------------------------
// Fused: cross-GEMM (f32 WMMA) -> argmin over 512 centroids -> atomic scatter.
// Protos tiles are double-buffered in LDS via CDNA5 async global->LDS copies
// (ASYNCcnt) so the tile nt+1 copy overlaps the 64 WMMAs of tile nt.
// ---------------------------------------------------------------------------
__global__ __launch_bounds__(256) void assign_scatter_kernel(
    const float* __restrict__ batch, const float* __restrict__ protos,
    const float* __restrict__ csq,
    float* __restrict__ out_sums, float* __restrict__ out_counts,
    float* __restrict__ out_closest)
{
    __shared__ __align__(16) float Ps[2][16 * PSTRIDE];  // double-buffered tile
    __shared__ float csq_s[Kv];

    const int tid   = threadIdx.x;
    const int lane  = tid & 31;
    const int wave  = tid >> 5;
    const int mrow  = lane & 15;          // A layout: M = lane % 16
    const int khalf = (lane >> 4) * 2;    // A layout: lanes>=16 hold K+2,K+3
    const int m0    = blockIdx.x * M_BLK + wave * 16;

    for (int i = tid; i < Kv; i += 256) csq_s[i] = csq[i];

    // Async-stage one 16x256 protos tile: 4 B128 DMAs per thread, no VGPR trip.
    auto stage_tile = [&](int nt, int buf) {
        #pragma unroll
        for (int i = 0; i < 4; ++i) {
            int f  = tid + 256 * i;       // float4 index 0..1023
            int r  = f >> 6;              // tile row 0..15
            int c4 = f & 63;              // float4 column
            const float* g = protos + (size_t)(nt * 16 + r) * Dv + c4 * 4;
            float*       l = &Ps[buf][r * PSTRIDE + c4 * 4];
            __builtin_amdgcn_global_load_async_to_lds_b128(
                (g_v4i*)g, (l_v4i*)l, 0, 0);
        }
    };

    // --- Load this wave's A fragments: X[m0+mrow][khalf + 4s .. +1], s=0..63 ---
    stage_tile(0, 0);                     // kick off tile 0 before the A loads
    v2f a[KSTEPS];
    const float* arow = batch + (size_t)(m0 + mrow) * Dv + khalf;
    #pragma unroll
    for (int s = 0; s < KSTEPS; ++s)
        a[s] = *(const v2f*)(arow + 4 * s);

    float minval[8];
    int   minidx[8];
    #pragma unroll
    for (int r = 0; r < 8; ++r) { minval[r] = 3.4e38f; minidx[r] = 0; }

    for (int nt = 0; nt < NTILES; ++nt) {
        const int cur = nt & 1;
        if (nt + 1 < NTILES) {
            stage_tile(nt + 1, cur ^ 1);              // overlap with compute
            __builtin_amdgcn_s_wait_asynccnt(4);      // tile nt landed (in-order)
        } else {
            __builtin_amdgcn_s_wait_asynccnt(0);
        }
        __syncthreads();                              // tile nt visible to all waves

        // B fragment: B[k][n] = P[n][k]; lane supplies (N=mrow, K=khalf+{0,1}).
        v8f acc0 = {}, acc1 = {};
        const float* brow = &Ps[cur][mrow * PSTRIDE + khalf];
        #pragma unroll
        for (int s = 0; s < KSTEPS; s += 2) {
            v2f b0 = *(const v2f*)(brow + 4 * s);
            v2f b1 = *(const v2f*)(brow + 4 * (s + 1));
            acc0 = __builtin_amdgcn_wmma_f32_16x16x4_f32(
                false, a[s],     false, b0, (short)0, acc0, false, false);
            acc1 = __builtin_amdgcn_wmma_f32_16x16x4_f32(
                false, a[s + 1], false, b1, (short)0, acc1, false, false);
        }

        // Score = ||c||^2 - 2*cross  (x_sq constant per row: argmin-invariant).
        float cq = csq_s[nt * 16 + mrow];
        int   n  = nt * 16 + mrow;
        #pragma unroll
        for (int r = 0; r < 8; ++r) {   // C layout: lane=N, VGPR r -> M=r(+8)
            float score = cq - 2.0f * (acc0[r] + acc1[r]);
            if (score < minval[r]) { minval[r] = score; minidx[r] = n; }
        }
        __syncthreads();   // all waves done reading Ps[cur] before it is re-staged
    }

    // --- Argmin butterfly across the 16 lanes sharing each row group. ---
    #pragma unroll
    for (int r = 0; r < 8; ++r) {
        float v = minval[r]; int ix = minidx[r];
        #pragma unroll
        for (int m = 8; m; m >>= 1) {   // masks < 16: stays within half-wave
            float ov = __shfl_xor(v, m, 32);
            int   oi = __shfl_xor(ix, m, 32);
            if (ov < v || (ov == v && oi < ix)) { v = ov; ix = oi; }
        }
        minidx[r] = ix;
    }
    // Exchange halves: lanes<16 now have rows 0-7, lanes>=16 rows 8-15.
    int other[8];
    #pragma unroll
    for (int r = 0; r < 8; ++r) other[r] = __shfl_xor(minidx[r], 16, 32);

    int closest;
    if (mrow < 8) closest = (lane < 16) ? minidx[mrow]     : other[mrow];
    else          closest = (lane < 16) ? other[mrow - 8]  : minidx[mrow - 8];

    if (lane < 16) {
        out_closest[m0 + mrow] = (float)closest;      // argmin index as float
        atomicAdd(out_counts + closest, 1.0f);        // one count per row
    }

    // --- Scatter this row into batchSums straight from the A registers. ---
    float* dst = out_sums + (size_t)closest * Dv + khalf;
    #pragma unroll
    for (int s = 0; s < KSTEPS; ++s) {
        atomicAdd(dst + 4 * s,     a[s].x);
        atomicAdd(dst + 4 * s + 1, a[s].y);
    }
}

// ---------------------------------------------------------------------------
// protos_new = (protoSums + batchSums) / max(protoCounts + counts, 1)
// ---------------------------------------------------------------------------
__global__ void finalize_kernel(const float* __restrict__ protoSums,
                                const float* __restrict__ protoCounts,
                                const float* __restrict__ batchSums,
                                const float* __restrict__ counts,
                                float* __restrict__ protos_new)
{
    int i = blockIdx.x * blockDim.x + threadIdx.x;
    if (i >= Kv * Dv) return;
    int k = i >> 8;                                   // i / Dv
    float ns = protoSums[i] + batchSums[i];
    float nc = protoCounts[k] + counts[k];
    protos_new[i] = ns / fmaxf(nc, 1.0f);
}

extern "C" void kernel_launch(void* const* d_in, const int* in_sizes, int n_in,
                              void* d_out, int out_size, void* d_ws, size_t ws_size,
                              hipStream_t stream) {
    (void)in_sizes; (void)n_in; (void)out_size; (void)ws_size;
    const float* batch       = (const float*)d_in[0];   // [64,1024,256]
    const float* protos      = (const float*)d_in[1];   // [512,256]
    const float* protoSums   = (const float*)d_in[2];   // [512,256]
    const float* protoCounts = (const float*)d_in[3];   // [512]

    float* out         = (float*)d_out;
    float* out_protos  = out;                       // [512*256]
    float* out_sums    = out + Kv * Dv;             // [512*256]
    float* out_counts  = out + 2 * Kv * Dv;         // [512]
    float* out_closest = out + 2 * Kv * Dv + Kv;    // [65536]
    float* csq         = (float*)d_ws;              // [512] scratch

    // Zero the atomic accumulator sections (d_out is poisoned, not re-zeroed).
    (void)hipMemsetAsync(out_sums, 0, (size_t)(Kv * Dv + Kv) * sizeof(float), stream);

    csq_kernel<<<Kv / 8, 256, 0, stream>>>(protos, csq);
    assign_scatter_kernel<<<Mv / M_BLK, 256, 0, stream>>>(
        batch, protos, csq, out_sums, out_counts, out_closest);
    finalize_kernel<<<(Kv * Dv + 255) / 256, 256, 0, stream>>>(
        protoSums, protoCounts, out_sums, out_counts, out_protos);
}